// DifferentiableTopK_29772713296402
// MI455X (gfx1250) — compile-verified
//
#include <hip/hip_runtime.h>
#include <hip/hip_bf16.h>

// DifferentiableTopK: out[i] = sigmoid((x[i] - kth)/TEMP), kth = (K+1)-th largest of x.
// N = 33,554,432 f32 (128 MB), K = 1024, TEMP = 0.1.
//
// Strategy (bandwidth-bound; MI455X 23.3 TB/s HBM, 192 MB L2 holds whole input):
//   pass1: 4096-bin LDS histogram of top 12 bits of order-preserving uint
//   pass2: 4096-bin histogram of next 12 bits, filtered by selected prefix
//   pass3: 256-bin histogram of low 8 bits, filtered by 24-bit prefix
//   small select kernels walk histograms descending -> exact kth float
//   pass4: fused sigmoid, NT loads (last use) + NT stores (streaming out)

typedef unsigned int u32;
typedef float v4f __attribute__((ext_vector_type(4)));

#define TOPK_K      1024
#define INV_TEMP    10.0f
#define NB_MAX      4096
#define NSUB        2           // sub-histograms per block (conflict reduction)
#define SUBSTRIDE   (NB_MAX + 4)  // +4 words -> bank skew between copies

// ---------------- workspace layout (u32 indices into d_ws) ----------------
#define WS_HIST1    0           // 4096
#define WS_HIST2    4096        // 4096
#define WS_HIST3    8192        // 256
#define WS_PREF1    8448        // 12-bit prefix after level 0
#define WS_K1       8449
#define WS_PREF2    8450        // 24-bit prefix after level 1
#define WS_K2       8451
#define WS_THRESH   8452        // kth value (float bits)
#define WS_WORDS    8453

__device__ __forceinline__ u32 f32_to_ord(float f) {
    u32 b = __float_as_uint(f);
    return b ^ ((b & 0x80000000u) ? 0xFFFFFFFFu : 0x80000000u);
}

__global__ void tk_init(u32* __restrict__ W) {
    int i = blockIdx.x * blockDim.x + threadIdx.x;
    if (i < WS_WORDS) W[i] = 0u;
}

// One histogram scan pass over the full array (grid-stride, B128 loads, RT hint
// so the array stays resident in the 192MB L2 for the following passes).
__global__ void tk_hist(const float* __restrict__ x, long long n,
                        u32* __restrict__ ghist, int nbins,
                        const u32* __restrict__ prefix_ptr, int prefix_shift,
                        int bin_shift, u32 bin_mask) {
    __shared__ u32 lh[NSUB * SUBSTRIDE];
    for (int i = threadIdx.x; i < NSUB * SUBSTRIDE; i += blockDim.x) lh[i] = 0u;
    __syncthreads();

    const bool use_pre = (prefix_ptr != nullptr);
    const u32  pre     = use_pre ? *prefix_ptr : 0u;
    u32* __restrict__ my = lh + (threadIdx.x & (NSUB - 1)) * SUBSTRIDE;

    const long long n4 = n >> 2;
    const long long gstride = (long long)gridDim.x * blockDim.x;
    const v4f* __restrict__ x4 = (const v4f*)x;

    for (long long i = (long long)blockIdx.x * blockDim.x + threadIdx.x; i < n4; i += gstride) {
        // speculative prefetch of next tile -> global_prefetch_b8 (safe if OOB)
        __builtin_prefetch((const void*)&x4[i + gstride], 0, 0);
        v4f v = x4[i];
#pragma unroll
        for (int c = 0; c < 4; ++c) {
            u32 u = f32_to_ord(v[c]);
            if (!use_pre || (u >> prefix_shift) == pre)
                atomicAdd(&my[(u >> bin_shift) & bin_mask], 1u);
        }
    }
    // scalar tail (n is 2^25 so normally empty; kept for robustness)
    if (blockIdx.x == 0 && threadIdx.x == 0) {
        for (long long i = (n4 << 2); i < n; ++i) {
            u32 u = f32_to_ord(x[i]);
            if (!use_pre || (u >> prefix_shift) == pre)
                atomicAdd(&my[(u >> bin_shift) & bin_mask], 1u);
        }
    }
    __syncthreads();
    for (int i = threadIdx.x; i < nbins; i += blockDim.x) {
        u32 c = 0;
#pragma unroll
        for (int s = 0; s < NSUB; ++s) c += lh[s * SUBSTRIDE + i];
        if (c) atomicAdd(&ghist[i], c);   // skip-zero: passes 2/3 flush almost nothing
    }
}

// Single-block select: find the bin (descending) containing the k-th largest.
// Writes new prefix = (prev<<bin_bits)|bin, remaining k, and (last level) the
// exact kth float reconstructed from the full 32-bit ordinal.
__global__ void tk_select(const u32* __restrict__ hist, int nbins,
                          const u32* __restrict__ k_in, u32 k_init,
                          const u32* __restrict__ prev_prefix, int bin_bits,
                          u32* __restrict__ out_prefix, u32* __restrict__ out_k,
                          float* __restrict__ out_thresh) {
    __shared__ u32 s[256];
    const int t   = threadIdx.x;
    const int bpt = nbins >> 8;          // bins per thread (16 or 1)
    const int base = t * bpt;

    u32 csum = 0;
    for (int i = 0; i < bpt; ++i) csum += hist[base + i];
    s[t] = csum;
    __syncthreads();
    // Hillis-Steele inclusive scan over 256 chunk sums
    for (int off = 1; off < 256; off <<= 1) {
        u32 add = (t >= off) ? s[t - off] : 0u;
        __syncthreads();
        s[t] += add;
        __syncthreads();
    }
    const u32 total = s[255];
    const u32 above = total - s[t];      // count in strictly-higher chunks
    const u32 k = k_in ? *k_in : k_init;

    if (above < k && k <= above + csum) {   // exactly one thread
        u32 c = above;
        for (int b = base + bpt - 1; b >= base; --b) {
            u32 h = hist[b];
            if (c + h >= k) {
                u32 np = prev_prefix ? (((*prev_prefix) << bin_bits) | (u32)b) : (u32)b;
                if (out_prefix) *out_prefix = np;
                if (out_k)      *out_k = k - c;
                if (out_thresh) {
                    u32 fb = (np & 0x80000000u) ? (np ^ 0x80000000u) : ~np;
                    *out_thresh = __uint_as_float(fb);
                }
                break;
            }
            c += h;
        }
    }
}

// Final fused pass: out = sigmoid((x - kth) * 10). NT load (last use of input),
// NT store (output never re-read) so we don't churn L2.
__global__ void tk_sigmoid(const float* __restrict__ x, float* __restrict__ out,
                           long long n, const float* __restrict__ thresh) {
    const float kth = *thresh;
    const float scale = INV_TEMP * 1.44269504088896340736f; // (1/T) * log2(e)
    const long long n4 = n >> 2;
    const long long gstride = (long long)gridDim.x * blockDim.x;
    const v4f* __restrict__ x4 = (const v4f*)x;
    v4f* __restrict__ o4 = (v4f*)out;

    for (long long i = (long long)blockIdx.x * blockDim.x + threadIdx.x; i < n4; i += gstride) {
        __builtin_prefetch((const void*)&x4[i + gstride], 0, 0);
        v4f v = __builtin_nontemporal_load(&x4[i]);
        v4f o;
#pragma unroll
        for (int c = 0; c < 4; ++c) {
            float e = exp2f((kth - v[c]) * scale);   // exp(-(x-kth)/T)
            o[c] = 1.0f / (1.0f + e);
        }
        __builtin_nontemporal_store(o, &o4[i]);
    }
    if (blockIdx.x == 0 && threadIdx.x == 0) {
        for (long long i = (n4 << 2); i < n; ++i) {
            float e = exp2f((kth - x[i]) * scale);
            out[i] = 1.0f / (1.0f + e);
        }
    }
}

extern "C" void kernel_launch(void* const* d_in, const int* in_sizes, int n_in,
                              void* d_out, int out_size, void* d_ws, size_t ws_size,
                              hipStream_t stream) {
    const float* x = (const float*)d_in[0];
    float* out = (float*)d_out;
    const long long n = (long long)in_sizes[0];

    u32* W = (u32*)d_ws;
    u32*   hist1 = W + WS_HIST1;
    u32*   hist2 = W + WS_HIST2;
    u32*   hist3 = W + WS_HIST3;
    u32*   pref1 = W + WS_PREF1;
    u32*   k1    = W + WS_K1;
    u32*   pref2 = W + WS_PREF2;
    u32*   k2    = W + WS_K2;
    float* thr   = (float*)(W + WS_THRESH);

    const int threads = 256;           // 8 waves (wave32)
    const int scanBlocks = 2048;
    const int sigBlocks  = 4096;
    const u32 kInit = (u32)(TOPK_K + 1);   // (K+1)-th largest

    tk_init<<<(WS_WORDS + 255) / 256, 256, 0, stream>>>(W);

    // level 0: top 12 bits
    tk_hist<<<scanBlocks, threads, 0, stream>>>(x, n, hist1, 4096,
                                                nullptr, 0, 20, 0xFFFu);
    tk_select<<<1, 256, 0, stream>>>(hist1, 4096, nullptr, kInit,
                                     nullptr, 12, pref1, k1, nullptr);
    // level 1: next 12 bits, prefix-filtered
    tk_hist<<<scanBlocks, threads, 0, stream>>>(x, n, hist2, 4096,
                                                pref1, 20, 8, 0xFFFu);
    tk_select<<<1, 256, 0, stream>>>(hist2, 4096, k1, 0u,
                                     pref1, 12, pref2, k2, nullptr);
    // level 2: low 8 bits, 24-bit prefix-filtered -> exact kth float
    tk_hist<<<scanBlocks, threads, 0, stream>>>(x, n, hist3, 256,
                                                pref2, 8, 0, 0xFFu);
    tk_select<<<1, 256, 0, stream>>>(hist3, 256, k2, 0u,
                                     pref2, 8, nullptr, nullptr, thr);
    // fused elementwise sigmoid
    tk_sigmoid<<<sigBlocks, threads, 0, stream>>>(x, out, n, thr);
}